// GCN_8856222564838
// MI455X (gfx1250) — compile-verified
//
#include <hip/hip_runtime.h>
#include <math.h>

// Problem constants (match reference)
#define NNODES 100000
#define NEDGES 1250000
#define FDIM   128
#define HDIM   64
#define CDIM   10
#define GDIM   1000

typedef float v2f __attribute__((ext_vector_type(2)));
typedef float v8f __attribute__((ext_vector_type(8)));

// ---------------------------------------------------------------- utilities
__global__ __launch_bounds__(256) void zero_f32(float* __restrict__ p, int n) {
  int i = blockIdx.x * blockDim.x + threadIdx.x;
  if (i < n) p[i] = 0.0f;
}

// ------------------------------------------------------------ degree kernels
__global__ __launch_bounds__(256) void deg_count(const int* __restrict__ dst,
                                                 float* __restrict__ deg, int e) {
  int i = blockIdx.x * blockDim.x + threadIdx.x;
  if (i < e) atomicAdd(&deg[dst[i]], 1.0f);
}

__global__ __launch_bounds__(256) void deg_finalize(float* __restrict__ deg, int n) {
  int i = blockIdx.x * blockDim.x + threadIdx.x;
  if (i < n) deg[i] = rsqrtf(deg[i] + 1.0f);   // +1 self loop
}

// --------------------------------------------------------------- WMMA GEMM
// Out[M x HDIM] = A[M x K] * B[K x HDIM], row-major, fp32 via V_WMMA_F32_16X16X4_F32.
// 8 waves / block, 1 16x16 output tile per wave. B staged in LDS (max 32 KB).
__global__ __launch_bounds__(256) void gemm_wmma_f32(const float* __restrict__ A,
                                                     const float* __restrict__ B,
                                                     float* __restrict__ Out,
                                                     int K, int ntilesN) {
  __shared__ float ldsB[FDIM * HDIM];           // 32 KB max (K<=128)
  for (int idx = threadIdx.x; idx < K * HDIM; idx += 256)
    ldsB[idx] = B[idx];
  __syncthreads();

  const int wave = threadIdx.x >> 5;
  const int lane = threadIdx.x & 31;
  const int tile = blockIdx.x * 8 + wave;
  const int tM = tile / ntilesN;
  const int tN = tile - tM * ntilesN;
  const int half = lane >> 4;                   // 0: K=k,k+1  1: K=k+2,k+3
  const int l    = lane & 15;

  const float* __restrict__ arow = A + (size_t)(tM * 16 + l) * K;
  const int col = tN * 16 + l;

  v8f c = {};
  for (int kk = 0; kk < K; kk += 4) {
    const int kb = kk + half * 2;
    v2f a, b;
    a.x = arow[kb];                             // contiguous pair -> b64 load
    a.y = arow[kb + 1];
    b.x = ldsB[kb * HDIM + col];
    b.y = ldsB[(kb + 1) * HDIM + col];
    // (neg_a, A, neg_b, B, c_mod, C, reuse_a, reuse_b)
    c = __builtin_amdgcn_wmma_f32_16x16x4_f32(false, a, false, b, (short)0, c,
                                              false, false);
  }

  // C/D layout: VGPR r -> M = r (lanes 0-15), M = r+8 (lanes 16-31); N = lane&15
  float* __restrict__ orow =
      Out + (size_t)(tM * 16 + half * 8) * HDIM + col;
#pragma unroll
  for (int r = 0; r < 8; ++r)
    orow[(size_t)r * HDIM] = c[r];
}

// ----------------------------------------------------------- edge scatter
// agg[dst] += h[src] * dinv[src]*dinv[dst]; 16 lanes per edge, float4 per lane.
__global__ __launch_bounds__(256) void scatter_add(const float* __restrict__ h,
                                                   const int* __restrict__ src,
                                                   const int* __restrict__ dst,
                                                   const float* __restrict__ dinv,
                                                   float* __restrict__ agg, int e) {
  int t = blockIdx.x * blockDim.x + threadIdx.x;
  int eidx = t >> 4;
  if (eidx >= e) return;
  int c = (t & 15) * 4;
  int s = src[eidx], d = dst[eidx];
  float ns = dinv[s] * dinv[d];
  float4 v = *reinterpret_cast<const float4*>(h + (size_t)s * HDIM + c);
  float* a = agg + (size_t)d * HDIM + c;
  atomicAdd(a + 0, v.x * ns);
  atomicAdd(a + 1, v.y * ns);
  atomicAdd(a + 2, v.z * ns);
  atomicAdd(a + 3, v.w * ns);
}

// ------------------------------------- layer-1 epilogue: self-loop + bias + relu
// agg <- relu(agg + h * dinv^2 + b)   (in place)
__global__ __launch_bounds__(256) void combine_relu(float* __restrict__ agg,
                                                    const float* __restrict__ h,
                                                    const float* __restrict__ dinv,
                                                    const float* __restrict__ bias,
                                                    int n) {
  int t = blockIdx.x * blockDim.x + threadIdx.x;
  int i = t >> 4;
  if (i >= n) return;
  int c = (t & 15) * 4;
  float d2 = dinv[i] * dinv[i];
  float4 av = *reinterpret_cast<const float4*>(agg + (size_t)i * HDIM + c);
  float4 hv = *reinterpret_cast<const float4*>(h + (size_t)i * HDIM + c);
  float4 o;
  o.x = fmaxf(av.x + hv.x * d2 + bias[c + 0], 0.0f);
  o.y = fmaxf(av.y + hv.y * d2 + bias[c + 1], 0.0f);
  o.z = fmaxf(av.z + hv.z * d2 + bias[c + 2], 0.0f);
  o.w = fmaxf(av.w + hv.w * d2 + bias[c + 3], 0.0f);
  *reinterpret_cast<float4*>(agg + (size_t)i * HDIM + c) = o;
}

// ------------- layer-2 epilogue fused with global mean-pool accumulation
__global__ __launch_bounds__(256) void combine_pool(const float* __restrict__ agg,
                                                    const float* __restrict__ h,
                                                    const float* __restrict__ dinv,
                                                    const float* __restrict__ bias,
                                                    const int* __restrict__ batch,
                                                    float* __restrict__ pooled,
                                                    float* __restrict__ cnt, int n) {
  int t = blockIdx.x * blockDim.x + threadIdx.x;
  int i = t >> 4;
  if (i >= n) return;
  int c = (t & 15) * 4;
  float d2 = dinv[i] * dinv[i];
  float4 av = *reinterpret_cast<const float4*>(agg + (size_t)i * HDIM + c);
  float4 hv = *reinterpret_cast<const float4*>(h + (size_t)i * HDIM + c);
  int g = batch[i];
  float* p = pooled + (size_t)g * HDIM + c;
  atomicAdd(p + 0, av.x + hv.x * d2 + bias[c + 0]);
  atomicAdd(p + 1, av.y + hv.y * d2 + bias[c + 1]);
  atomicAdd(p + 2, av.z + hv.z * d2 + bias[c + 2]);
  atomicAdd(p + 3, av.w + hv.w * d2 + bias[c + 3]);
  if (c == 0) atomicAdd(&cnt[g], 1.0f);
}

// ------------------------- head: mean, logits = pooled@Wl + bl, log_softmax
__global__ __launch_bounds__(256) void head_kernel(const float* __restrict__ pooled,
                                                   const float* __restrict__ cnt,
                                                   const float* __restrict__ Wl,
                                                   const float* __restrict__ bl,
                                                   float* __restrict__ out, int g) {
  int gi = blockIdx.x * blockDim.x + threadIdx.x;
  if (gi >= g) return;
  float inv = 1.0f / fmaxf(cnt[gi], 1.0f);
  const float* prow = pooled + (size_t)gi * HDIM;
  float z[CDIM];
  float mx = -1e30f;
#pragma unroll
  for (int c = 0; c < CDIM; ++c) {
    float acc = bl[c];
    for (int j = 0; j < HDIM; ++j)
      acc += prow[j] * inv * Wl[j * CDIM + c];
    z[c] = acc;
    mx = fmaxf(mx, acc);
  }
  float se = 0.0f;
#pragma unroll
  for (int c = 0; c < CDIM; ++c) se += __expf(z[c] - mx);
  float lse = mx + __logf(se);
#pragma unroll
  for (int c = 0; c < CDIM; ++c) out[(size_t)gi * CDIM + c] = z[c] - lse;
}

// ---------------------------------------------------------------- launcher
extern "C" void kernel_launch(void* const* d_in, const int* in_sizes, int n_in,
                              void* d_out, int out_size, void* d_ws, size_t ws_size,
                              hipStream_t stream) {
  const float* x   = (const float*)d_in[0];
  const int*   ei  = (const int*)d_in[1];     // [2, E] flat
  const int*   src = ei;
  const int*   dst = ei + NEDGES;
  const int*   batch = (const int*)d_in[2];
  const float* W1 = (const float*)d_in[3];
  const float* b1 = (const float*)d_in[4];
  const float* W2 = (const float*)d_in[5];
  const float* b2 = (const float*)d_in[6];
  const float* Wl = (const float*)d_in[7];
  const float* bl = (const float*)d_in[8];
  float* out = (float*)d_out;

  // carve workspace (256-B aligned)
  char* ws = (char*)d_ws;
  size_t off = 0;
  auto carve = [&](size_t bytes) {
    char* p = ws + off;
    off = (off + bytes + 255) & ~(size_t)255;
    return p;
  };
  float* dinv   = (float*)carve((size_t)NNODES * 4);
  float* hbuf   = (float*)carve((size_t)NNODES * HDIM * 4);
  float* aggbuf = (float*)carve((size_t)NNODES * HDIM * 4);
  float* pooled = (float*)carve((size_t)GDIM * HDIM * 4);
  float* cnt    = (float*)carve((size_t)GDIM * 4);

  const int ZT = 256;
  const int gemm_blocks = (NNODES / 16) * (HDIM / 16) / 8;   // 3125
  const int scat_blocks = (NEDGES * 16) / 256;               // 78125
  const int comb_blocks = (NNODES * 16) / 256;               // 6250

  // degrees
  zero_f32<<<(NNODES + ZT - 1) / ZT, ZT, 0, stream>>>(dinv, NNODES);
  deg_count<<<(NEDGES + ZT - 1) / ZT, ZT, 0, stream>>>(dst, dinv, NEDGES);
  deg_finalize<<<(NNODES + ZT - 1) / ZT, ZT, 0, stream>>>(dinv, NNODES);

  // layer 1: h = x @ W1 (fp32 WMMA), scatter, self-loop + bias + relu
  gemm_wmma_f32<<<gemm_blocks, 256, 0, stream>>>(x, W1, hbuf, FDIM, HDIM / 16);
  zero_f32<<<(NNODES * HDIM + ZT - 1) / ZT, ZT, 0, stream>>>(aggbuf, NNODES * HDIM);
  scatter_add<<<scat_blocks, 256, 0, stream>>>(hbuf, src, dst, dinv, aggbuf, NEDGES);
  combine_relu<<<comb_blocks, 256, 0, stream>>>(aggbuf, hbuf, dinv, b1, NNODES);

  // layer 2: h2 = h1' @ W2, scatter, epilogue fused with mean-pool accumulation
  gemm_wmma_f32<<<gemm_blocks, 256, 0, stream>>>(aggbuf, W2, hbuf, HDIM, HDIM / 16);
  zero_f32<<<(NNODES * HDIM + ZT - 1) / ZT, ZT, 0, stream>>>(aggbuf, NNODES * HDIM);
  scatter_add<<<scat_blocks, 256, 0, stream>>>(hbuf, src, dst, dinv, aggbuf, NEDGES);

  zero_f32<<<(GDIM * HDIM + ZT - 1) / ZT, ZT, 0, stream>>>(pooled, GDIM * HDIM);
  zero_f32<<<(GDIM + ZT - 1) / ZT, ZT, 0, stream>>>(cnt, GDIM);
  combine_pool<<<comb_blocks, 256, 0, stream>>>(aggbuf, hbuf, dinv, b2, batch,
                                                pooled, cnt, NNODES);

  // head + log_softmax
  head_kernel<<<(GDIM + ZT - 1) / ZT, ZT, 0, stream>>>(pooled, cnt, Wl, bl, out, GDIM);
}